// Net_81295140979034
// MI455X (gfx1250) — compile-verified
//
#include <hip/hip_runtime.h>
#include <math.h>

typedef __attribute__((ext_vector_type(2))) float v2f;
typedef __attribute__((ext_vector_type(8))) float v8f;

#define DIMS 50
#define KWIN 51
#define EE 6
#define BB 8
#define MAXK 3456  // largest Ci*27 = 128*27

// ---------------------------------------------------------------------------
// Kernel 0: precompute 1D Gaussian rows g[e][51], per-channel scale a_e/S,
// and zero the atomic-max cell.  S = sum_e a_e * (sum_k g_e[k])^3 reproduces
// the reference's normalization by the GLOBAL kernel sum.
// ---------------------------------------------------------------------------
__global__ void prep_kernel(const float* __restrict__ sigma,
                            float* __restrict__ gw,
                            float* __restrict__ cscale,
                            unsigned int* __restrict__ maxbits) {
  int tid = threadIdx.x;
  for (int i = tid; i < EE * KWIN; i += blockDim.x) {
    int e = i / KWIN, k = i % KWIN;
    float d   = (float)k - (float)(KWIN - 1) * 0.5f;
    float var = sigma[e] * sigma[e];
    gw[i] = expf(-(d * d) / (2.0f * var));
  }
  __syncthreads();
  if (tid == 0) {
    float S = 0.0f;
    float amp[EE];
    for (int e = 0; e < EE; ++e) {
      float var = sigma[e] * sigma[e];
      amp[e] = 1.0f / (2.0f * 3.14159265358979323846f * var);
      float s = 0.0f;
      for (int k = 0; k < KWIN; ++k) s += gw[e * KWIN + k];
      S += amp[e] * s * s * s;
    }
    for (int e = 0; e < EE; ++e) cscale[e] = amp[e] / S;
    *maxbits = 0u;
  }
}

// ---------------------------------------------------------------------------
// Kernel 1: one separable 1D Gaussian pass along an axis with stride `st`
// (elements). Zero padding via bounds check (pad=25 in reference).
// finalpass: multiply by a_e/S and fold a block-reduce + atomicMax of the
// output (all values >= 0, so float bit pattern is order preserving).
// ---------------------------------------------------------------------------
__global__ void blur_pass(const float* __restrict__ in, float* __restrict__ outp,
                          const float* __restrict__ gw,
                          const float* __restrict__ cscale,
                          unsigned int* __restrict__ maxbits,
                          int st, int finalpass) {
  __shared__ float g[KWIN];
  __shared__ float red[256];
  const int be = blockIdx.y;       // b*E + e
  const int e  = be % EE;
  if (threadIdx.x < KWIN) g[threadIdx.x] = gw[e * KWIN + threadIdx.x];
  __syncthreads();

  const int CH = DIMS * DIMS * DIMS;  // 125000
  int f = blockIdx.x * blockDim.x + threadIdx.x;
  float val = 0.0f;
  if (f < CH) {
    int c = (f / st) % DIMS;          // coordinate along the filtered axis
    size_t base = (size_t)be * CH;
    const float* ip = in + base + f;
    float s = 0.0f;
    #pragma unroll 1
    for (int k = 0; k < KWIN; ++k) {
      int cc = c + k - (KWIN - 1) / 2;
      if (cc >= 0 && cc < DIMS) s += g[k] * ip[(k - (KWIN - 1) / 2) * st];
    }
    if (finalpass) s *= cscale[e];
    outp[base + f] = s;
    val = s;
  }
  if (finalpass) {
    red[threadIdx.x] = val;
    __syncthreads();
    for (int off = 128; off > 0; off >>= 1) {
      if ((int)threadIdx.x < off)
        red[threadIdx.x] = fmaxf(red[threadIdx.x], red[threadIdx.x + off]);
      __syncthreads();
    }
    if (threadIdx.x == 0) atomicMax(maxbits, __float_as_uint(red[0]));
  }
}

// ---------------------------------------------------------------------------
// Kernel 2: 3x3x3 VALID conv as implicit GEMM on V_WMMA_F32_16X16X4_F32.
//   Each wave computes a 32(co) x 16(position) tile: TWO 16x16 WMMA
//   accumulators sharing one B fragment (the scattered input gathers),
//   doubling matrix-op density per byte gathered.
//   K = Ci*27; main loop guard-free, unrolled 4x (16 K / iter, 8 WMMAs);
//   guarded tail only for conv1 (K=162).  im2col offsets precomputed in LDS.
// A frag: lane l -> A[m = l%16][k = kb + 2*(l/16) + j]   (j = vgpr 0/1)
// B frag: lane l -> B[k = kb + 2*(l/16) + j][n = l%16]
// D frag: lane l, vgpr j -> D[m = 8*(l/16) + j][n = l%16]
// use_invmax folds the reference's  xc / max(xc)  into conv1.
// ---------------------------------------------------------------------------
__global__ void conv_wmma(const float* __restrict__ in,
                          const float* __restrict__ wgt,
                          const float* __restrict__ bias,
                          float* __restrict__ outp,
                          const float* __restrict__ maxptr, int use_invmax,
                          int Ci, int Co, int Si, int So) {
  __shared__ int tbl[MAXK];
  const int K = Ci * 27;
  const int P = So * So * So;

  // im2col offset table (incremental decode: blockDim 128 = 4*27 + 20)
  {
    int kk = threadIdx.x;
    int ci = kk / 27;
    int r  = kk - ci * 27;
    while (kk < K) {
      int dz = r / 9, dy = (r / 3) % 3, dx = r - (r / 3) * 3;
      tbl[kk] = ((ci * Si + dz) * Si + dy) * Si + dx;
      kk += 128; ci += 4; r += 20;
      if (r >= 27) { r -= 27; ci += 1; }
    }
  }
  __syncthreads();

  const int wave   = threadIdx.x >> 5;
  const int lane   = threadIdx.x & 31;
  const int lane16 = lane & 15;
  const int half   = lane >> 4;

  const int posTile = (blockIdx.x * 4 + wave) * 16;
  const int coBase  = blockIdx.y * 32;     // 32 output channels per wave
  const int img     = blockIdx.z;

  const int p  = posTile + lane16;
  const int pc = (p < P) ? p : (P - 1);
  const int oz = pc / (So * So);
  const int r2 = pc - oz * So * So;
  const int oy = r2 / So;
  const int ox = r2 - oy * So;

  const float* ib  = in + (size_t)img * Ci * Si * Si * Si
                        + ((size_t)oz * Si + oy) * Si + ox;
  const float* wr0 = wgt + (size_t)(coBase + lane16) * K;
  const float* wr1 = wr0 + (size_t)16 * K;

  v8f acc0 = {};
  v8f acc1 = {};

  // ---- main body: K rounded down to a multiple of 16, no guards ----
  const int Kmain = K & ~15;
  #pragma unroll 1
  for (int kb = 0; kb < Kmain; kb += 16) {
    #pragma unroll
    for (int u = 0; u < 4; ++u) {
      const int k0 = kb + 4 * u + 2 * half;          // even
      const v2f a0 = *(const v2f*)(wr0 + k0);        // global_load_b64
      const v2f a1 = *(const v2f*)(wr1 + k0);        // global_load_b64
      const int2 t = *(const int2*)(tbl + k0);       // ds_load_b64
      v2f b;
      b.x = ib[t.x];
      b.y = ib[t.y];
      acc0 = __builtin_amdgcn_wmma_f32_16x16x4_f32(
          false, a0, false, b, (short)0, acc0, false, false);
      acc1 = __builtin_amdgcn_wmma_f32_16x16x4_f32(
          false, a1, false, b, (short)0, acc1, false, false);
    }
  }

  // ---- guarded tail (only conv1: K=162 -> kb=160) ----
  #pragma unroll 1
  for (int kb = Kmain; kb < K; kb += 4) {
    int k0 = kb + 2 * half;
    int k1 = k0 + 1;
    v2f a0, a1, b;
    a0.x = 0.0f; a0.y = 0.0f; a1.x = 0.0f; a1.y = 0.0f;
    b.x = 0.0f; b.y = 0.0f;
    if (k0 < K) { a0.x = wr0[k0]; a1.x = wr1[k0]; b.x = ib[tbl[k0]]; }
    if (k1 < K) { a0.y = wr0[k1]; a1.y = wr1[k1]; b.y = ib[tbl[k1]]; }
    acc0 = __builtin_amdgcn_wmma_f32_16x16x4_f32(
        false, a0, false, b, (short)0, acc0, false, false);
    acc1 = __builtin_amdgcn_wmma_f32_16x16x4_f32(
        false, a1, false, b, (short)0, acc1, false, false);
  }

  float scale = 1.0f;
  if (use_invmax) scale = 1.0f / (*maxptr);

  if (p < P) {
    #pragma unroll
    for (int j = 0; j < 8; ++j) {
      int co0 = coBase + 8 * half + j;
      int co1 = co0 + 16;
      outp[((size_t)img * Co + co0) * P + p] = acc0[j] * scale + bias[co0];
      outp[((size_t)img * Co + co1) * P + p] = acc1[j] * scale + bias[co1];
    }
  }
}

// ---------------------------------------------------------------------------
// Kernel 3: 2x2x2 max-pool (VALID, floor) followed by relu.
// ---------------------------------------------------------------------------
__global__ void pool_relu(const float* __restrict__ in, float* __restrict__ outp,
                          int C, int Sc, int Sp, int total) {
  int o = blockIdx.x * blockDim.x + threadIdx.x;
  if (o >= total) return;
  int t  = o;
  int xp = t % Sp; t /= Sp;
  int yp = t % Sp; t /= Sp;
  int zp = t % Sp; t /= Sp;
  int c  = t % C;
  int b  = t / C;
  const float* ib = in + (((size_t)(b * C + c) * Sc + 2 * zp) * Sc + 2 * yp) * Sc + 2 * xp;
  float m = -INFINITY;
  #pragma unroll
  for (int dz = 0; dz < 2; ++dz)
    #pragma unroll
    for (int dy = 0; dy < 2; ++dy)
      #pragma unroll
      for (int dx = 0; dx < 2; ++dx)
        m = fmaxf(m, ib[(dz * Sc + dy) * Sc + dx]);
  outp[o] = fmaxf(m, 0.0f);
}

// ---------------------------------------------------------------------------
// Kernel 4: FC head  (8,256) -> relu(fc 128) -> fc 29.  Tiny; single block.
// ---------------------------------------------------------------------------
__global__ void fc_head(const float* __restrict__ v,
                        const float* __restrict__ fw1, const float* __restrict__ fb1,
                        const float* __restrict__ fw2, const float* __restrict__ fb2,
                        float* __restrict__ outp) {
  __shared__ float vs[BB * 256];
  __shared__ float y1[BB * 128];
  for (int i = threadIdx.x; i < BB * 256; i += blockDim.x) vs[i] = v[i];
  __syncthreads();
  for (int i = threadIdx.x; i < BB * 128; i += blockDim.x) {
    int b = i / 128, j = i % 128;
    float s = fb1[j];
    for (int k = 0; k < 256; ++k) s += vs[b * 256 + k] * fw1[j * 256 + k];
    y1[i] = fmaxf(s, 0.0f);
  }
  __syncthreads();
  for (int i = threadIdx.x; i < BB * 29; i += blockDim.x) {
    int b = i / 29, t = i % 29;
    float s = fb2[t];
    for (int j = 0; j < 128; ++j) s += y1[b * 128 + j] * fw2[t * 128 + j];
    outp[i] = s;
  }
}

// ---------------------------------------------------------------------------
static inline int cdiv(int a, int b) { return (a + b - 1) / b; }

extern "C" void kernel_launch(void* const* d_in, const int* in_sizes, int n_in,
                              void* d_out, int out_size, void* d_ws, size_t ws_size,
                              hipStream_t stream) {
  (void)in_sizes; (void)n_in; (void)out_size; (void)ws_size;

  const float* x     = (const float*)d_in[0];
  const float* sigma = (const float*)d_in[1];
  const float* w1 = (const float*)d_in[2];  const float* b1 = (const float*)d_in[3];
  const float* w2 = (const float*)d_in[4];  const float* b2 = (const float*)d_in[5];
  const float* w3 = (const float*)d_in[6];  const float* b3 = (const float*)d_in[7];
  const float* w4 = (const float*)d_in[8];  const float* b4 = (const float*)d_in[9];
  const float* fw1 = (const float*)d_in[10]; const float* fb1 = (const float*)d_in[11];
  const float* fw2 = (const float*)d_in[12]; const float* fb2 = (const float*)d_in[13];
  float* outp = (float*)d_out;

  // ---- workspace layout (256B aligned offsets) ----
  char* w = (char*)d_ws;
  float*        gw      = (float*)(w + 0);          // 6*51 floats
  float*        cscale  = (float*)(w + 1536);       // 6 floats
  unsigned int* maxbits = (unsigned int*)(w + 1792);// 1 uint (float bits)
  const float*  maxf    = (const float*)(w + 1792);
  float* tA      = (float*)(w + 4096);                        // 6,000,000 f
  float* tB      = (float*)(w + 4096 + 24000000);             // 6,000,000 f
  float* convbuf = (float*)(w + 4096 + 48000000);             // up to 28,311,552 f
  float* poolbuf = (float*)(w + 4096 + 48000000 + 113246208); // up to 3,538,944 f

  const int CH = DIMS * DIMS * DIMS;  // 125000 per (b,e) channel

  // 0) Gaussian rows + scales + zero the max cell
  prep_kernel<<<1, 128, 0, stream>>>(sigma, gw, cscale, maxbits);

  // 1) separable blur: z, y, x (last pass applies a_e/S and computes max)
  dim3 bgrid(cdiv(CH, 256), BB * EE);
  blur_pass<<<bgrid, 256, 0, stream>>>(x,  tA, gw, cscale, maxbits, DIMS * DIMS, 0);
  blur_pass<<<bgrid, 256, 0, stream>>>(tA, tB, gw, cscale, maxbits, DIMS,        0);
  blur_pass<<<bgrid, 256, 0, stream>>>(tB, tA, gw, cscale, maxbits, 1,           1);

  // 2) conv stack (WMMA implicit GEMM, 32co x 16pos per wave) + pool/relu
  // conv1: 6->32, 50->48, /max folded in
  conv_wmma<<<dim3(cdiv(48*48*48, 64), 32/32, BB), 128, 0, stream>>>(
      tA, w1, b1, convbuf, maxf, 1, 6, 32, 50, 48);
  { int tot = BB*32*24*24*24;
    pool_relu<<<cdiv(tot,256), 256, 0, stream>>>(convbuf, poolbuf, 32, 48, 24, tot); }

  // conv2: 32->64, 24->22
  conv_wmma<<<dim3(cdiv(22*22*22, 64), 64/32, BB), 128, 0, stream>>>(
      poolbuf, w2, b2, convbuf, maxf, 0, 32, 64, 24, 22);
  { int tot = BB*64*11*11*11;
    pool_relu<<<cdiv(tot,256), 256, 0, stream>>>(convbuf, poolbuf, 64, 22, 11, tot); }

  // conv3: 64->128, 11->9
  conv_wmma<<<dim3(cdiv(9*9*9, 64), 128/32, BB), 128, 0, stream>>>(
      poolbuf, w3, b3, convbuf, maxf, 0, 64, 128, 11, 9);
  { int tot = BB*128*4*4*4;
    pool_relu<<<cdiv(tot,256), 256, 0, stream>>>(convbuf, poolbuf, 128, 9, 4, tot); }

  // conv4: 128->256, 4->2
  conv_wmma<<<dim3(cdiv(2*2*2, 64), 256/32, BB), 128, 0, stream>>>(
      poolbuf, w4, b4, convbuf, maxf, 0, 128, 256, 4, 2);
  { int tot = BB*256;
    pool_relu<<<cdiv(tot,256), 256, 0, stream>>>(convbuf, poolbuf, 256, 2, 1, tot); }

  // 3) FC head -> (8,29)
  fc_head<<<1, 256, 0, stream>>>(poolbuf, fw1, fb1, fw2, fb2, outp);
}